// VolumeRendererModule_1675037245903
// MI455X (gfx1250) — compile-verified
//
#include <hip/hip_runtime.h>
#include <cstdint>

#define GRID_N    128
#define BLOCK     256
#define LOOKAHEAD 8

typedef __attribute__((ext_vector_type(4))) unsigned int v4u_t;
typedef __attribute__((ext_vector_type(8))) int          v8i_t;
typedef __attribute__((ext_vector_type(4))) int          v4i_t;

__device__ __forceinline__ int clampg(int v) {
  return v < 0 ? 0 : (v > GRID_N - 1 ? GRID_N - 1 : v);
}

__global__ __launch_bounds__(BLOCK) void volume_render_fwd_kernel(
    const float* __restrict__ w_sigma,   // [128,128,128]
    const float* __restrict__ rays,      // [n,6]
    float* __restrict__ out,             // [n,4]
    int n_rays)
{
  __shared__ float sRays[BLOCK * 6];     // 6 KB of ray data for this block
  const int tid  = threadIdx.x;
  const long long base = (long long)blockIdx.x * BLOCK;

#if __has_builtin(__builtin_amdgcn_tensor_load_to_lds)
  // --- Stage ray data into LDS with the Tensor Data Mover (one wave issues).
  if (tid < 32) {
    const uint64_t ga    = (uint64_t)(uintptr_t)(rays + base * 6);
    const uint32_t la    = (uint32_t)(uintptr_t)(&sRays[0]); // low 32b = LDS offset
    const uint32_t elems = BLOCK * 6;                        // 1536 f32 tile
    // D# group 0: count=1 | lds_addr | global_addr[56:0] | type=2 ("image")
    v4u_t g0 = { 1u, la, (uint32_t)ga,
                 (uint32_t)((ga >> 32) & 0x01FFFFFFu) | (2u << 30) };
    // D# group 1: data_size=4B; tensor_dim0=1536; tensor_dim1=1;
    //             tile_dim0=1536; tile_dim1=1; tensor_dim0_stride=1536
    v8i_t g1 = { (int)(2u << 16),
                 (int)((elems & 0xFFFFu) << 16),   // tensor_dim0[15:0] @ bits 63:48
                 (int)(1u << 16),                  // dim0 hi=0 | tensor_dim1 lo=1
                 (int)((elems & 0xFFFFu) << 16),   // dim1 hi=0 | tile_dim0
                 1,                                // tile_dim1=1, tile_dim2=0
                 (int)elems,                       // tensor_dim0_stride[31:0]
                 0, 0 };
    v4i_t gz = { 0, 0, 0, 0 };
#if defined(__clang_major__) && __clang_major__ >= 23
    v8i_t z8 = { 0, 0, 0, 0, 0, 0, 0, 0 };
    __builtin_amdgcn_tensor_load_to_lds(g0, g1, gz, gz, z8, 0);
#else
    __builtin_amdgcn_tensor_load_to_lds(g0, g1, gz, gz, 0);
#endif
    __builtin_amdgcn_s_wait_tensorcnt(0);
  }
#else
  for (int i = tid; i < BLOCK * 6; i += BLOCK) sRays[i] = rays[base * 6 + i];
#endif
  __syncthreads();

  const int ray = (int)base + tid;
  if (ray >= n_rays) return;

  const float* rp = &sRays[tid * 6];
  const float ox = rp[0], oy = rp[1], oz = rp[2];
  const float rx = rp[3], ry = rp[4], rz = rp[5];

  // Normalize direction (ray_aabb uses normalized dn).
  const float rn = 1.0f / sqrtf(rx * rx + ry * ry + rz * rz);
  const float dx = rx * rn, dy = ry * rn, dz = rz * rn;

  // Ray/AABB slab test against [-1.5, 1.5]^3.
  const float qx = 1.0f / (dx == 0.0f ? 1e-20f : dx);
  const float qy = 1.0f / (dy == 0.0f ? 1e-20f : dy);
  const float qz = 1.0f / (dz == 0.0f ? 1e-20f : dz);
  float a, b;
  a = (-1.5f - ox) * qx; b = (1.5f - ox) * qx;
  const float nx = fminf(a, b), fx = fmaxf(a, b);
  a = (-1.5f - oy) * qy; b = (1.5f - oy) * qy;
  const float ny = fminf(a, b), fy = fmaxf(a, b);
  a = (-1.5f - oz) * qz; b = (1.5f - oz) * qz;
  const float nz = fminf(a, b), fz = fmaxf(a, b);

  float nearv = fmaxf(fmaxf(nx, ny), nz);
  const float farv = fminf(fminf(fx, fy), fz);
  const bool hit = (nearv <= farv);
  nearv = hit ? nearv : 0.0f;
  const float span = hit ? (farv - nearv) : 0.0f;
  const int   ns   = hit ? (int)fminf(span * 32.0f, 256.0f) : 0;
  const float nsf  = (float)(ns > 1 ? ns : 1);
  const float dist = span / nsf;    // step length (== dt per sample)

  float T = 1.0f;   // running transmittance cumprod(1 - alpha)
  float C = 1.0f;   // running product of (1 + alpha * T)

  for (int j = 0; j < ns; ++j) {
    const float t  = nearv + dist * ((float)j + 0.5f);
    const int ix = clampg((int)floorf(((ox + dx * t) * (1.0f / 3.0f) + 0.5f) * (float)GRID_N));
    const int iy = clampg((int)floorf(((oy + dy * t) * (1.0f / 3.0f) + 0.5f) * (float)GRID_N));
    const int iz = clampg((int)floorf(((oz + dz * t) * (1.0f / 3.0f) + 0.5f) * (float)GRID_N));
    const int idx = (ix * GRID_N + iy) * GRID_N + iz;

    // Look-ahead prefetch of the sigma gather stream (address is data-independent).
    if (j + LOOKAHEAD < ns) {
      const float tp  = nearv + dist * ((float)(j + LOOKAHEAD) + 0.5f);
      const int px = clampg((int)floorf(((ox + dx * tp) * (1.0f / 3.0f) + 0.5f) * (float)GRID_N));
      const int py = clampg((int)floorf(((oy + dy * tp) * (1.0f / 3.0f) + 0.5f) * (float)GRID_N));
      const int pz = clampg((int)floorf(((oz + dz * tp) * (1.0f / 3.0f) + 0.5f) * (float)GRID_N));
      __builtin_prefetch(w_sigma + ((px * GRID_N + py) * GRID_N + pz), 0, 3);
    }

    const float sg    = fmaxf(w_sigma[idx], 0.0f);
    const float alpha = 1.0f - __expf(sg * dist);
    T *= (1.0f - alpha);
    C  = fmaf(alpha * T, C, C);     // C *= (1 + alpha * T)
  }

  float4 res;
  res.x = C; res.y = C; res.z = C; res.w = 1.0f - C;
  reinterpret_cast<float4*>(out)[ray] = res;
}

extern "C" void kernel_launch(void* const* d_in, const int* in_sizes, int n_in,
                              void* d_out, int out_size, void* d_ws, size_t ws_size,
                              hipStream_t stream) {
  const float* w_sigma = (const float*)d_in[0];
  // d_in[1] (w_rgb) is provably dead in the reference: it only enters the
  // output as `c + 0.0 * sum(relu(finite))` == c + 0.0. We elide ~3.6 GB of
  // random gathers by never touching it.
  const float* rays = (const float*)d_in[2];
  const int n_rays = in_sizes[2] / 6;
  const int grid   = (n_rays + BLOCK - 1) / BLOCK;
  volume_render_fwd_kernel<<<grid, BLOCK, 0, stream>>>(
      w_sigma, rays, (float*)d_out, n_rays);
  (void)d_ws; (void)ws_size; (void)n_in; (void)out_size;
}